// Attn_72842645340585
// MI455X (gfx1250) — compile-verified
//
#include <hip/hip_runtime.h>

typedef __attribute__((ext_vector_type(2))) float v2f;
typedef __attribute__((ext_vector_type(8))) float v8f;

#define HDIM 2048
#define SDIM 8192

#define ROWS    128            // rows per workgroup tile (8 waves x 16)
#define BK      32             // K columns staged per chunk
#define LSTRIDE (BK + 4)       // padded LDS row stride (floats): banks 36*m%64 all distinct
#define KSPLIT  4
#define KCHUNK  (HDIM / KSPLIT)        // 512 K per block
#define NCHUNK  (KCHUNK / BK)          // 16 stages

// ---------------------------------------------------------------------------
// Zero-init scratch: v (HDIM floats) followed by energies (SDIM floats)
// ---------------------------------------------------------------------------
__global__ void __launch_bounds__(256) attn_zero_ws(float* ws, int n) {
    int i = blockIdx.x * 256 + threadIdx.x;
    if (i < n) ws[i] = 0.0f;
}

// ---------------------------------------------------------------------------
// v[k] = sum_j W[j][k] * h[j]   (v = W^T h). Fully coalesced column reads,
// h[j] uniform (scalar loads). K-split partials combined with fp32 atomics.
// ---------------------------------------------------------------------------
__global__ void __launch_bounds__(256) attn_wT_h(const float* __restrict__ W,
                                                 const float* __restrict__ h,
                                                 float* __restrict__ v) {
    const int col = blockIdx.x * 256 + threadIdx.x;
    const int j0  = blockIdx.y * 128;
    const float* wp = W + (size_t)j0 * HDIM + col;
    float acc = 0.0f;
#pragma unroll 8
    for (int j = 0; j < 128; ++j)
        acc = fmaf(wp[(size_t)j * HDIM], h[j0 + j], acc);
    atomicAdd(&v[col], acc);
}

// ---------------------------------------------------------------------------
// energies = enc(SxH) @ v(H) via V_WMMA_F32_16X16X4_F32, with enc staged
// through LDS using gfx1250 async loads (coalesced, double-buffered).
// ---------------------------------------------------------------------------
__device__ __forceinline__ unsigned lds_off(const void* p) {
    // generic pointer to LDS: low 32 bits are the in-wave LDS byte offset
    return (unsigned)(uintptr_t)p;
}

__device__ __forceinline__ void stage_chunk(const float* __restrict__ enc,
                                            float* __restrict__ buf,
                                            int row0, int kstart, int t) {
    const int cg = (t & 7) * 4;        // float4 column within chunk
    const int rs = t >> 3;             // 0..31
#pragma unroll
    for (int i = 0; i < 4; ++i) {
        const int r = rs + 32 * i;
        const unsigned voff   = (unsigned)(((row0 + r) * HDIM + kstart + cg) * 4);
        const unsigned dsaddr = lds_off(buf + r * LSTRIDE + cg);
        asm volatile("global_load_async_to_lds_b128 %0, %1, %2"
                     :: "v"(dsaddr), "v"(voff), "s"(enc)
                     : "memory");
    }
}

__global__ void __launch_bounds__(256) attn_energies_wmma(
    const float* __restrict__ enc, const float* __restrict__ v,
    float* __restrict__ energies) {
    __shared__ __align__(16) float tile[2][ROWS * LSTRIDE];   // 2 x 4608 floats
    __shared__ __align__(16) float vbuf[KCHUNK];              // 512 floats

    const int t       = threadIdx.x;
    const int lane    = t & 31;
    const int wave    = t >> 5;
    const int rowbase = blockIdx.x * ROWS;
    const int kbase   = blockIdx.y * KCHUNK;

    // stage this block's slice of v into LDS (coalesced)
    vbuf[t]       = v[kbase + t];
    vbuf[t + 256] = v[kbase + t + 256];

    // prologue: stage chunk 0
    stage_chunk(enc, tile[0], rowbase, kbase, t);
    asm volatile("s_wait_asynccnt 0x0" ::: "memory");
    __syncthreads();

    // WMMA fragment mapping (ISA 7.12.2, 32-bit A 16x4 / B 4x16):
    // lanes 0-15 -> K = 0,1 ; lanes 16-31 -> K = 2,3 ; A row = lane & 15.
    const int m   = lane & 15;
    const int klo = (lane < 16) ? 0 : 2;
    const int lr  = wave * 16 + m;     // local row owned by this lane

    v8f c = {};
    for (int ch = 0; ch < NCHUNK; ++ch) {
        const int cur = ch & 1;
        if (ch + 1 < NCHUNK)           // prefetch next chunk into other buffer
            stage_chunk(enc, tile[cur ^ 1], rowbase, kbase + (ch + 1) * BK, t);

        const float* abase = &tile[cur][lr * LSTRIDE + klo];
        const float* bbase = &vbuf[ch * BK + klo];
#pragma unroll
        for (int kk = 0; kk < BK; kk += 4) {
            v2f a = *(const v2f*)(abase + kk);   // ds_load_b64, conflict-free
            v2f b = *(const v2f*)(bbase + kk);   // ds broadcast
            c = __builtin_amdgcn_wmma_f32_16x16x4_f32(
                false, a, false, b, (short)0, c, false, false);
        }

        asm volatile("s_wait_asynccnt 0x0" ::: "memory");
        __syncthreads();
    }

    // D column 0: rows 0..7 -> lane 0 VGPR 0..7 ; rows 8..15 -> lane 16.
    if (lane == 0 || lane == 16) {
        const int roff = rowbase + wave * 16 + ((lane == 16) ? 8 : 0);
#pragma unroll
        for (int i = 0; i < 8; ++i)
            atomicAdd(&energies[roff + i], c[i]);
    }
}

// ---------------------------------------------------------------------------
// Softmax over SDIM energies. Single 1024-thread block, LDS reductions.
// ---------------------------------------------------------------------------
__global__ void __launch_bounds__(1024) attn_softmax(
    const float* __restrict__ e, float* __restrict__ out) {
    __shared__ float red[1024];
    const int t = threadIdx.x;

    float m = -3.0e38f;
    for (int i = t; i < SDIM; i += 1024) m = fmaxf(m, e[i]);
    red[t] = m; __syncthreads();
    for (int s = 512; s > 0; s >>= 1) {
        if (t < s) red[t] = fmaxf(red[t], red[t + s]);
        __syncthreads();
    }
    const float gmax = red[0];
    __syncthreads();

    float sum = 0.0f;
    for (int i = t; i < SDIM; i += 1024) {
        float x = __expf(e[i] - gmax);
        out[i] = x;                    // cache exp() in output buffer
        sum += x;
    }
    red[t] = sum; __syncthreads();
    for (int s = 512; s > 0; s >>= 1) {
        if (t < s) red[t] += red[t + s];
        __syncthreads();
    }
    const float inv = 1.0f / red[0];

    for (int i = t; i < SDIM; i += 1024)
        out[i] *= inv;
}

// ---------------------------------------------------------------------------
// Inputs: 0=hidden(1,1,H) 1=encoder_outputs(S,1,H) 2=W(H,H) 3=b(H), fp32.
// Output: attn(1,1,S) fp32.  b is unused: b.h is a softmax-invariant shift.
// ---------------------------------------------------------------------------
extern "C" void kernel_launch(void* const* d_in, const int* in_sizes, int n_in,
                              void* d_out, int out_size, void* d_ws, size_t ws_size,
                              hipStream_t stream) {
    const float* hidden = (const float*)d_in[0];
    const float* enc    = (const float*)d_in[1];
    const float* W      = (const float*)d_in[2];

    float* v        = (float*)d_ws;          // HDIM floats
    float* energies = v + HDIM;              // SDIM floats
    float* out      = (float*)d_out;

    const int nzero = HDIM + SDIM;
    attn_zero_ws<<<(nzero + 255) / 256, 256, 0, stream>>>((float*)d_ws, nzero);

    attn_wT_h<<<dim3(HDIM / 256, HDIM / 128), 256, 0, stream>>>(W, hidden, v);

    attn_energies_wmma<<<dim3(SDIM / ROWS, KSPLIT), 256, 0, stream>>>(enc, v, energies);

    attn_softmax<<<1, 1024, 0, stream>>>(energies, out);
}